// GraphNetworkLayer_68461778698666
// MI455X (gfx1250) — compile-verified
//
#include <hip/hip_runtime.h>

typedef __attribute__((ext_vector_type(2))) float v2f;
typedef __attribute__((ext_vector_type(8))) float v8f;

#define B_ 8
#define N_ 4096
#define E_ 16384
#define G_ 64
#define D_ 128

// ---- d_out layout (4-byte elements, reference tuple order) ----
#define OFF_NODES 0
#define LEN_NODES (B_*N_*D_)
#define OFF_EDGES (OFF_NODES + LEN_NODES)
#define LEN_EDGES (B_*E_*D_)
#define OFF_RECV  (OFF_EDGES + LEN_EDGES)
#define LEN_BE_   (B_*E_)
#define OFF_SEND  (OFF_RECV + LEN_BE_)
#define OFF_GLOB  (OFF_SEND + LEN_BE_)
#define LEN_GLOB  (B_*G_*D_)
#define OFF_NGI   (OFF_GLOB + LEN_GLOB)
#define LEN_BN_   (B_*N_)
#define OFF_EGI   (OFF_NGI + LEN_BN_)

// ---- workspace layout (floats) ----
#define WS_ADJ   0                        // [B,N,D] scatter of edges_update by receiver
#define WS_NAGG  (WS_ADJ + B_*N_*D_)      // [B,G,D] scatter of nodes_update
#define WS_EAGG  (WS_NAGG + B_*G_*D_)     // [B,G,D] scatter of edges_update
#define WS_TOTAL (WS_EAGG + B_*G_*D_)

// =============== prep: zero scatter buffers + passthrough ints ===============
__global__ __launch_bounds__(256) void gn_prep(float* __restrict__ ws, int* __restrict__ oi,
                                               const int* __restrict__ recv, const int* __restrict__ send,
                                               const int* __restrict__ ngi, const int* __restrict__ egi) {
    long long i = (long long)blockIdx.x * blockDim.x + threadIdx.x;
    long long stride = (long long)gridDim.x * blockDim.x;
    for (long long j = i; j < (long long)WS_TOTAL; j += stride) ws[j] = 0.0f;
    for (long long j = i; j < (long long)B_*E_; j += stride) {
        oi[OFF_RECV + j] = recv[j];
        oi[OFF_SEND + j] = send[j];
        oi[OFF_EGI  + j] = egi[j];
    }
    for (long long j = i; j < (long long)B_*N_; j += stride) oi[OFF_NGI + j] = ngi[j];
}

// =============== 16x16 tile GEMM via V_WMMA_F32_16X16X4_F32 ===============
// A_lds: 16 x KTOT tile, row stride STRIDE floats (padded, conflict-free).
// Bw:    [KTOT,128] row-major weights in global (L2-resident).
// ncol:  this wave's 16-column base within D=128.
template<int KTOT, int STRIDE>
__device__ inline v8f tile_gemm(const float* __restrict__ A_lds,
                                const float* __restrict__ Bw, int ncol) {
    const int lane = threadIdx.x & 31;
    const int half = lane >> 4;   // lanes 16..31 hold K+2/K+3 (A) and rows K+2/K+3 (B)
    const int l16  = lane & 15;
    const float* arow = A_lds + l16 * STRIDE + 2 * half;          // A: M=l16
    const float* bcol = Bw + 2 * half * D_ + ncol + l16;          // B: N=ncol+l16
    v8f acc = {};
    #pragma unroll 4
    for (int k = 0; k < KTOT; k += 4) {
        v2f a; a.x = arow[k]; a.y = arow[k + 1];
        v2f b; b.x = bcol[k * D_]; b.y = bcol[k * D_ + D_];
        acc = __builtin_amdgcn_wmma_f32_16x16x4_f32(false, a, false, b,
                                                    (short)0, acc, false, false);
    }
    return acc;
}

// =============== edge block: [E,512]@We -> relu; fused residual + 2 scatters ===============
__global__ __launch_bounds__(256) void gn_edge(
        const float* __restrict__ nodes, const float* __restrict__ edges,
        const int* __restrict__ recv, const int* __restrict__ send,
        const float* __restrict__ glob, const int* __restrict__ egi,
        const float* __restrict__ We, const float* __restrict__ be,
        const float* __restrict__ wres_p,
        float* __restrict__ out, float* __restrict__ ws) {
    constexpr int K   = 4 * D_;   // 512
    constexpr int STR = K + 4;    // pad -> bank-conflict-free column reads
    __shared__ float A_lds[16 * STR];
    __shared__ int ridx[16], gidx[16];

    const int et = blockIdx.x;    // edge tile (16 rows)
    const int b  = blockIdx.y;
    const int t  = threadIdx.x;

    if (t < 16) {
        const int e = et * 16 + t;
        ridx[t] = recv[b * E_ + e];
        gidx[t] = egi[b * E_ + e];
    }
    {   // stage A = [edges | sent | recv | glob] : 16 rows x 16 chunks of 32 floats
        const int r = t >> 4, chunk = t & 15;
        const int e = et * 16 + r;
        const int seg = chunk >> 2, c0 = (chunk & 3) * 32;
        const float* src;
        if (seg == 0)      src = edges + ((size_t)b * E_ + e) * D_ + c0;
        else if (seg == 1) src = nodes + ((size_t)b * N_ + send[b * E_ + e]) * D_ + c0;
        else if (seg == 2) src = nodes + ((size_t)b * N_ + recv[b * E_ + e]) * D_ + c0;
        else               src = glob  + ((size_t)b * G_ + egi[b * E_ + e]) * D_ + c0;
        float* dst = A_lds + r * STR + chunk * 32;
        #pragma unroll
        for (int j = 0; j < 32; j += 4) *(float4*)(dst + j) = *(const float4*)(src + j);
    }
    __syncthreads();

    const int ncol = (t >> 5) * 16;
    v8f acc = tile_gemm<K, STR>(A_lds, We, ncol);

    const int lane = t & 31, half = lane >> 4, l16 = lane & 15;
    const int c = ncol + l16;
    const float bias = be[c];
    const float wres = wres_p[0];
    float* edges_out = out + OFF_EDGES;
    float* adj  = ws + WS_ADJ;
    float* eagg = ws + WS_EAGG;
    #pragma unroll
    for (int i = 0; i < 8; ++i) {
        const int r = 8 * half + i;            // C/D layout: VGPR i -> M = 8*half + i
        const int e = et * 16 + r;
        float u = acc[i] + bias;
        u = u > 0.0f ? u : 0.0f;               // relu -> edges_update
        const size_t o = ((size_t)b * E_ + e) * D_ + c;
        edges_out[o] = edges[o] + wres * u;    // residual
        atomicAdd(adj  + ((size_t)b * N_ + ridx[r]) * D_ + c, u);  // segment_sum by receiver
        atomicAdd(eagg + ((size_t)b * G_ + gidx[r]) * D_ + c, u);  // segment_sum by edge graph
    }
}

// =============== node block: [N,384]@Wn -> relu; fused residual + scatter ===============
__global__ __launch_bounds__(256) void gn_node(
        const float* __restrict__ nodes, const float* __restrict__ glob,
        const int* __restrict__ ngi,
        const float* __restrict__ Wn, const float* __restrict__ bn,
        const float* __restrict__ wres_p,
        float* __restrict__ out, float* __restrict__ ws) {
    constexpr int K   = 3 * D_;   // 384
    constexpr int STR = K + 4;
    __shared__ float A_lds[16 * STR];
    __shared__ int gdx[16];

    const int nt = blockIdx.x;
    const int b  = blockIdx.y;
    const int t  = threadIdx.x;
    const float* adj = ws + WS_ADJ;

    if (t < 16) gdx[t] = ngi[b * N_ + nt * 16 + t];
    if (t < 192) {  // 16 rows x 12 chunks of 32 floats
        const int r = t / 12, chunk = t % 12;
        const int n = nt * 16 + r;
        const int seg = chunk >> 2, c0 = (chunk & 3) * 32;
        const float* src;
        if (seg == 0)      src = nodes + ((size_t)b * N_ + n) * D_ + c0;
        else if (seg == 1) src = adj   + ((size_t)b * N_ + n) * D_ + c0;
        else               src = glob  + ((size_t)b * G_ + ngi[b * N_ + n]) * D_ + c0;
        float* dst = A_lds + r * STR + chunk * 32;
        #pragma unroll
        for (int j = 0; j < 32; j += 4) *(float4*)(dst + j) = *(const float4*)(src + j);
    }
    __syncthreads();

    const int ncol = (t >> 5) * 16;
    v8f acc = tile_gemm<K, STR>(A_lds, Wn, ncol);

    const int lane = t & 31, half = lane >> 4, l16 = lane & 15;
    const int c = ncol + l16;
    const float bias = bn[c];
    const float wres = wres_p[0];
    float* nodes_out = out + OFF_NODES;
    float* nagg = ws + WS_NAGG;
    #pragma unroll
    for (int i = 0; i < 8; ++i) {
        const int r = 8 * half + i;
        const int n = nt * 16 + r;
        float u = acc[i] + bias;
        u = u > 0.0f ? u : 0.0f;
        const size_t o = ((size_t)b * N_ + n) * D_ + c;
        nodes_out[o] = nodes[o] + wres * u;
        atomicAdd(nagg + ((size_t)b * G_ + gdx[r]) * D_ + c, u);   // segment_sum by node graph
    }
}

// =============== global block: [G,384]@Wg -> relu; fused residual ===============
__global__ __launch_bounds__(256) void gn_glob(
        const float* __restrict__ glob,
        const float* __restrict__ Wg, const float* __restrict__ bg,
        const float* __restrict__ wres_p,
        float* __restrict__ out, float* __restrict__ ws) {
    constexpr int K   = 3 * D_;
    constexpr int STR = K + 4;
    __shared__ float A_lds[16 * STR];

    const int gt = blockIdx.x;
    const int b  = blockIdx.y;
    const int t  = threadIdx.x;
    const float* nagg = ws + WS_NAGG;
    const float* eagg = ws + WS_EAGG;

    if (t < 192) {
        const int r = t / 12, chunk = t % 12;
        const int g = gt * 16 + r;
        const int seg = chunk >> 2, c0 = (chunk & 3) * 32;
        const float* src;
        if (seg == 0)      src = nagg + ((size_t)b * G_ + g) * D_ + c0;
        else if (seg == 1) src = eagg + ((size_t)b * G_ + g) * D_ + c0;
        else               src = glob + ((size_t)b * G_ + g) * D_ + c0;
        float* dst = A_lds + r * STR + chunk * 32;
        #pragma unroll
        for (int j = 0; j < 32; j += 4) *(float4*)(dst + j) = *(const float4*)(src + j);
    }
    __syncthreads();

    const int ncol = (t >> 5) * 16;
    v8f acc = tile_gemm<K, STR>(A_lds, Wg, ncol);

    const int lane = t & 31, half = lane >> 4, l16 = lane & 15;
    const int c = ncol + l16;
    const float bias = bg[c];
    const float wres = wres_p[0];
    float* glob_out = out + OFF_GLOB;
    #pragma unroll
    for (int i = 0; i < 8; ++i) {
        const int r = 8 * half + i;
        const int g = gt * 16 + r;
        float u = acc[i] + bias;
        u = u > 0.0f ? u : 0.0f;
        const size_t o = ((size_t)b * G_ + g) * D_ + c;
        glob_out[o] = glob[o] + wres * u;
    }
}

extern "C" void kernel_launch(void* const* d_in, const int* in_sizes, int n_in,
                              void* d_out, int out_size, void* d_ws, size_t ws_size,
                              hipStream_t stream) {
    const float* nodes = (const float*)d_in[0];
    const float* edges = (const float*)d_in[1];
    const int*   recv  = (const int*)  d_in[2];
    const int*   send  = (const int*)  d_in[3];
    const float* glob  = (const float*)d_in[4];
    const int*   ngi   = (const int*)  d_in[5];
    const int*   egi   = (const int*)  d_in[6];
    const float* We    = (const float*)d_in[7];
    const float* be    = (const float*)d_in[8];
    const float* Wn    = (const float*)d_in[9];
    const float* bn    = (const float*)d_in[10];
    const float* Wg    = (const float*)d_in[11];
    const float* bg    = (const float*)d_in[12];
    const float* wres  = (const float*)d_in[13];
    float* out = (float*)d_out;
    float* ws  = (float*)d_ws;

    gn_prep<<<dim3(1024), dim3(256), 0, stream>>>(ws, (int*)out, recv, send, ngi, egi);
    gn_edge<<<dim3(E_ / 16, B_), dim3(256), 0, stream>>>(nodes, edges, recv, send, glob, egi,
                                                         We, be, wres, out, ws);
    gn_node<<<dim3(N_ / 16, B_), dim3(256), 0, stream>>>(nodes, glob, ngi, Wn, bn, wres, out, ws);
    gn_glob<<<dim3(G_ / 16, B_), dim3(256), 0, stream>>>(glob, Wg, bg, wres, out, ws);
}